// OptimalTransportAlignment_43413529428593
// MI455X (gfx1250) — compile-verified
//
#include <hip/hip_runtime.h>
#include <hip/hip_bf16.h>
#include <math.h>

// ---------------------------------------------------------------------------
// OptimalTransportAlignment for MI455X (gfx1250, wave32, WMMA).
//
// Pipeline:
//   1) sqA/sqB row norms
//   2) Gram via V_WMMA_F32_16X16X4_F32 -> dist = sqrt(max(.,0)) clamp, global max
//   3) logK = -max(dist/maxd, 1e-6) / reg   (in-place, fp32, 256 MB in ws)
//   4) 200x { g = -lse_col(logK + f); f = -lse_row(logK + g) }  (online LSE,
//      HBM-bound: ~512 MB/iter -> ~4.4 ms floor at 23.3 TB/s; all streaming
//      loads are b128 so the loop stays bandwidth- not issue-limited)
//   5) aligned_A = T @ A, aligned_B = T^T @ B with T = exp(logK+f+g) computed
//      on the fly inside fp32 WMMA GEMMs (16x64 macro-tile per wave).
// ---------------------------------------------------------------------------

typedef __attribute__((ext_vector_type(2))) float v2f;
typedef __attribute__((ext_vector_type(8))) float v8f;

#define N_PTS 8192
#define D_DIM 768
#define REG_F 0.01f
#define INV_REG_F 100.0f
#define MAX_ITER 200
#define ROW_CHUNKS 64
#define CHUNK_ROWS (N_PTS / ROW_CHUNKS)   // 128
#define NEG_LOG_N (-9.0109130249f)        // -log(8192)

// Online logsumexp update: (m, s) <- merge with element x.
__device__ __forceinline__ void lse_update(float& m, float& s, float x) {
    const float nm = fmaxf(m, x);
    s = s * __expf(m - nm) + __expf(x - nm);
    m = nm;
}
// Merge two online stats.
__device__ __forceinline__ void lse_merge(float& m, float& s, float m2, float s2) {
    const float nm = fmaxf(m, m2);
    s = s * __expf(m - nm) + s2 * __expf(m2 - nm);
    m = nm;
}

// ---------------- small utility kernels ----------------

__global__ void __launch_bounds__(256)
ota_rowsq_kernel(const float* __restrict__ X, float* __restrict__ sq) {
    const int lane = threadIdx.x & 31;
    const int wave = threadIdx.x >> 5;
    const int row  = blockIdx.x * 8 + wave;
    const float* xr = X + (size_t)row * D_DIM;
    float s = 0.f;
    for (int k = lane; k < D_DIM; k += 32) { float v = xr[k]; s += v * v; }
    for (int off = 16; off > 0; off >>= 1) s += __shfl_xor(s, off, 32);
    if (lane == 0) sq[row] = s;
}

__global__ void __launch_bounds__(256)
ota_init_kernel(float* __restrict__ f, unsigned int* __restrict__ maxbits) {
    const int i = blockIdx.x * 256 + threadIdx.x;
    if (i < N_PTS) f[i] = NEG_LOG_N;
    if (i == 0) *maxbits = 0u;   // all dist >= 0, uint order == float order
}

// ---------------- stage 2: Gram + dist via fp32 WMMA ----------------
// One wave computes a 16x64 tile of dist (4 accumulators reuse the A-frag).

__global__ void __launch_bounds__(256)
ota_gram_dist_kernel(const float* __restrict__ A, const float* __restrict__ B,
                     const float* __restrict__ sqA, const float* __restrict__ sqB,
                     float* __restrict__ dist, unsigned int* __restrict__ maxbits) {
    const int lane = threadIdx.x & 31;
    const int wave = threadIdx.x >> 5;
    const int task = blockIdx.x * 8 + wave;
    const int NMAC = N_PTS / 64;               // 128 macro-n tiles
    const int tm = task / NMAC;
    const int tn = task % NMAC;
    const int i0 = tm * 16;
    const int j0 = tn * 64;
    const int mr   = lane & 15;
    const int koff = (lane >> 4) * 2;          // A/B frag: lanes 16-31 hold K=2,3

    const float* arow = A + (size_t)(i0 + mr) * D_DIM;
    const float* brow[4];
    #pragma unroll
    for (int nt = 0; nt < 4; ++nt)
        brow[nt] = B + (size_t)(j0 + nt * 16 + mr) * D_DIM;

    v8f acc[4] = {v8f{}, v8f{}, v8f{}, v8f{}};
    for (int kb = 0; kb < D_DIM; kb += 4) {
        v2f a;
        a.x = arow[kb + koff];
        a.y = arow[kb + koff + 1];
        #pragma unroll
        for (int nt = 0; nt < 4; ++nt) {
            v2f b;
            b.x = brow[nt][kb + koff];
            b.y = brow[nt][kb + koff + 1];
            acc[nt] = __builtin_amdgcn_wmma_f32_16x16x4_f32(
                false, a, false, b, (short)0, acc[nt], false, false);
        }
    }

    // C/D layout: VGPR v, lanes 0-15 -> M=v, lanes 16-31 -> M=v+8; N = lane&15
    const int ci = i0 + (lane >> 4) * 8;
    float lmax = 0.f;
    #pragma unroll
    for (int nt = 0; nt < 4; ++nt) {
        const int j = j0 + nt * 16 + (lane & 15);
        const float sbj = sqB[j];
        #pragma unroll
        for (int v = 0; v < 8; ++v) {
            const int i = ci + v;
            float d = sqA[i] + sbj - 2.0f * acc[nt][v];
            d = sqrtf(fmaxf(d, 0.0f));
            d = fmaxf(d, 1e-6f);
            dist[(size_t)i * N_PTS + j] = d;
            lmax = fmaxf(lmax, d);
        }
    }
    for (int off = 16; off > 0; off >>= 1) lmax = fmaxf(lmax, __shfl_xor(lmax, off, 32));
    if (lane == 0) atomicMax(maxbits, __float_as_uint(lmax));
}

// ---------------- stage 3: dist -> logK in place ----------------

__global__ void __launch_bounds__(256)
ota_scale_kernel(float* __restrict__ logK, const unsigned int* __restrict__ maxbits) {
    const float inv = 1.0f / __uint_as_float(*maxbits);
    const size_t idx = ((size_t)blockIdx.x * 256 + threadIdx.x) * 4;
    float4 v = *(const float4*)(logK + idx);
    v.x = -fmaxf(v.x * inv, 1e-6f) * INV_REG_F;
    v.y = -fmaxf(v.y * inv, 1e-6f) * INV_REG_F;
    v.z = -fmaxf(v.z * inv, 1e-6f) * INV_REG_F;
    v.w = -fmaxf(v.w * inv, 1e-6f) * INV_REG_F;
    *(float4*)(logK + idx) = v;
}

// ---------------- stage 4: sinkhorn (online logsumexp, b128 streaming) -----
// Column LSE: each thread owns 4 consecutive columns; partial over 128-row
// chunks (grid (8,64) = 512 blocks = 4096 waves in flight), then combine.

__global__ void __launch_bounds__(256)
ota_col_lse_partial(const float* __restrict__ logK, const float* __restrict__ f,
                    float* __restrict__ pm, float* __restrict__ ps) {
    const int j0 = (blockIdx.x * 256 + threadIdx.x) * 4;  // 4 columns / thread
    const int c  = blockIdx.y;                            // row chunk
    const int i0 = c * CHUNK_ROWS;
    const float* p = logK + (size_t)i0 * N_PTS + j0;
    float m0 = -INFINITY, m1 = -INFINITY, m2 = -INFINITY, m3 = -INFINITY;
    float s0 = 0.f, s1 = 0.f, s2 = 0.f, s3 = 0.f;
    #pragma unroll 4
    for (int r = 0; r < CHUNK_ROWS; ++r) {
        if (r + 4 < CHUNK_ROWS)
            __builtin_prefetch(p + (size_t)(r + 4) * N_PTS, 0, 3);
        const float fr = f[i0 + r];                       // uniform -> scalar
        const float4 x = *(const float4*)(p + (size_t)r * N_PTS);
        lse_update(m0, s0, x.x + fr);
        lse_update(m1, s1, x.y + fr);
        lse_update(m2, s2, x.z + fr);
        lse_update(m3, s3, x.w + fr);
    }
    *(float4*)(pm + (size_t)c * N_PTS + j0) = make_float4(m0, m1, m2, m3);
    *(float4*)(ps + (size_t)c * N_PTS + j0) = make_float4(s0, s1, s2, s3);
}

__global__ void __launch_bounds__(256)
ota_col_lse_combine(const float* __restrict__ pm, const float* __restrict__ ps,
                    float* __restrict__ g) {
    const int j0 = (blockIdx.x * 256 + threadIdx.x) * 4;
    float m0 = -INFINITY, m1 = -INFINITY, m2 = -INFINITY, m3 = -INFINITY;
    float s0 = 0.f, s1 = 0.f, s2 = 0.f, s3 = 0.f;
    for (int c = 0; c < ROW_CHUNKS; ++c) {
        const float4 mm = *(const float4*)(pm + (size_t)c * N_PTS + j0);
        const float4 sv = *(const float4*)(ps + (size_t)c * N_PTS + j0);
        lse_merge(m0, s0, mm.x, sv.x);
        lse_merge(m1, s1, mm.y, sv.y);
        lse_merge(m2, s2, mm.z, sv.z);
        lse_merge(m3, s3, mm.w, sv.w);
    }
    *(float4*)(g + j0) = make_float4(-(m0 + __logf(s0)), -(m1 + __logf(s1)),
                                     -(m2 + __logf(s2)), -(m3 + __logf(s3)));
}

// Row LSE: one 256-thread block per row, b128 loads, shuffle+LDS (m,s) merge.
__global__ void __launch_bounds__(256)
ota_row_lse_kernel(const float* __restrict__ logK, const float* __restrict__ g,
                   float* __restrict__ f) {
    __shared__ float sm[8], ss[8];
    const int i = blockIdx.x;
    const int tid = threadIdx.x, lane = tid & 31, wave = tid >> 5;
    const float* row = logK + (size_t)i * N_PTS;
    float m = -INFINITY, s = 0.0f;
    for (int j = tid * 4; j < N_PTS; j += 1024) {
        if (j + 4096 < N_PTS) __builtin_prefetch(row + j + 4096, 0, 3);
        const float4 x  = *(const float4*)(row + j);
        const float4 gv = *(const float4*)(g + j);
        lse_update(m, s, x.x + gv.x);
        lse_update(m, s, x.y + gv.y);
        lse_update(m, s, x.z + gv.z);
        lse_update(m, s, x.w + gv.w);
    }
    for (int off = 16; off > 0; off >>= 1) {
        const float m2 = __shfl_xor(m, off, 32);
        const float s2 = __shfl_xor(s, off, 32);
        lse_merge(m, s, m2, s2);
    }
    if (lane == 0) { sm[wave] = m; ss[wave] = s; }
    __syncthreads();
    if (tid == 0) {
        float M = sm[0], S = ss[0];
        #pragma unroll
        for (int w = 1; w < 8; ++w) lse_merge(M, S, sm[w], ss[w]);
        f[i] = -(M + __logf(S));
    }
}

// ---------------- stage 5: T @ A and T^T @ B via fp32 WMMA ----------------
// T[i,j] = exp(logK[i,j] + f[i] + g[j]) computed on the fly; 16x64 tile/wave.

__global__ void __launch_bounds__(256)
ota_apply_T_A(const float* __restrict__ logK, const float* __restrict__ A,
              const float* __restrict__ f, const float* __restrict__ g,
              float* __restrict__ out) {
    const int lane = threadIdx.x & 31;
    const int wave = threadIdx.x >> 5;
    const int task = blockIdx.x * 8 + wave;
    const int NMAC = D_DIM / 64;               // 12
    const int tm = task / NMAC;
    const int tn = task % NMAC;
    const int i0 = tm * 16, d0 = tn * 64;
    const int mr   = lane & 15;
    const int koff = (lane >> 4) * 2;
    const int i    = i0 + mr;                  // A-frag row
    const float fi = f[i];
    const float* lrow = logK + (size_t)i * N_PTS;

    v8f acc[4] = {v8f{}, v8f{}, v8f{}, v8f{}};
    for (int jb = 0; jb < N_PTS; jb += 4) {
        const int jj = jb + koff;
        v2f a;
        a.x = __expf(lrow[jj]     + fi + g[jj]);
        a.y = __expf(lrow[jj + 1] + fi + g[jj + 1]);
        const float* br = A + (size_t)jj * D_DIM + d0;   // rows jj, jj+1 of A
        #pragma unroll
        for (int nt = 0; nt < 4; ++nt) {
            v2f b;
            b.x = br[nt * 16 + (lane & 15)];
            b.y = br[D_DIM + nt * 16 + (lane & 15)];
            acc[nt] = __builtin_amdgcn_wmma_f32_16x16x4_f32(
                false, a, false, b, (short)0, acc[nt], false, false);
        }
    }
    const int ci = i0 + (lane >> 4) * 8;
    #pragma unroll
    for (int nt = 0; nt < 4; ++nt)
        #pragma unroll
        for (int v = 0; v < 8; ++v)
            out[(size_t)(ci + v) * D_DIM + d0 + nt * 16 + (lane & 15)] = acc[nt][v];
}

__global__ void __launch_bounds__(256)
ota_apply_T_B(const float* __restrict__ logK, const float* __restrict__ B,
              const float* __restrict__ f, const float* __restrict__ g,
              float* __restrict__ out) {
    const int lane = threadIdx.x & 31;
    const int wave = threadIdx.x >> 5;
    const int task = blockIdx.x * 8 + wave;
    const int NMAC = D_DIM / 64;
    const int tm = task / NMAC;
    const int tn = task % NMAC;
    const int j0 = tm * 16, d0 = tn * 64;
    const int mr   = lane & 15;
    const int koff = (lane >> 4) * 2;
    const int j    = j0 + mr;                  // A-frag row = output column j
    const float gj = g[j];

    v8f acc[4] = {v8f{}, v8f{}, v8f{}, v8f{}};
    for (int ib = 0; ib < N_PTS; ib += 4) {
        const int ii = ib + koff;
        v2f a;
        a.x = __expf(logK[(size_t)ii * N_PTS + j]       + f[ii]     + gj);
        a.y = __expf(logK[(size_t)(ii + 1) * N_PTS + j] + f[ii + 1] + gj);
        const float* br = B + (size_t)ii * D_DIM + d0;  // rows ii, ii+1 of B
        #pragma unroll
        for (int nt = 0; nt < 4; ++nt) {
            v2f b;
            b.x = br[nt * 16 + (lane & 15)];
            b.y = br[D_DIM + nt * 16 + (lane & 15)];
            acc[nt] = __builtin_amdgcn_wmma_f32_16x16x4_f32(
                false, a, false, b, (short)0, acc[nt], false, false);
        }
    }
    const int cj = j0 + (lane >> 4) * 8;
    #pragma unroll
    for (int nt = 0; nt < 4; ++nt)
        #pragma unroll
        for (int v = 0; v < 8; ++v)
            out[(size_t)(cj + v) * D_DIM + d0 + nt * 16 + (lane & 15)] = acc[nt][v];
}

// ---------------- host launcher ----------------

extern "C" void kernel_launch(void* const* d_in, const int* in_sizes, int n_in,
                              void* d_out, int out_size, void* d_ws, size_t ws_size,
                              hipStream_t stream) {
    (void)in_sizes; (void)n_in; (void)out_size; (void)ws_size;
    const float* A = (const float*)d_in[0];
    const float* B = (const float*)d_in[1];
    float* out = (float*)d_out;

    // workspace layout
    char* ws = (char*)d_ws;
    float* logK = (float*)ws;                                  // N*N fp32 (256 MB)
    size_t off = (size_t)N_PTS * N_PTS * sizeof(float);
    float* pm = (float*)(ws + off); off += (size_t)ROW_CHUNKS * N_PTS * sizeof(float);
    float* ps = (float*)(ws + off); off += (size_t)ROW_CHUNKS * N_PTS * sizeof(float);
    float* sqA = (float*)(ws + off); off += N_PTS * sizeof(float);
    float* sqB = (float*)(ws + off); off += N_PTS * sizeof(float);
    float* f   = (float*)(ws + off); off += N_PTS * sizeof(float);
    float* g   = (float*)(ws + off); off += N_PTS * sizeof(float);
    unsigned int* maxbits = (unsigned int*)(ws + off);

    // 1) row norms + init
    ota_rowsq_kernel<<<N_PTS / 8, 256, 0, stream>>>(A, sqA);
    ota_rowsq_kernel<<<N_PTS / 8, 256, 0, stream>>>(B, sqB);
    ota_init_kernel<<<N_PTS / 256, 256, 0, stream>>>(f, maxbits);

    // 2) Gram -> dist (+ global max) via WMMA: 512 m-tiles x 128 macro-n tiles
    ota_gram_dist_kernel<<<(512 * 128) / 8, 256, 0, stream>>>(A, B, sqA, sqB,
                                                              logK, maxbits);
    // 3) dist -> logK
    ota_scale_kernel<<<(int)((size_t)N_PTS * N_PTS / 1024), 256, 0, stream>>>(logK, maxbits);

    // 4) sinkhorn iterations (g first, then f — matches reference ordering)
    dim3 gp(N_PTS / 1024, ROW_CHUNKS);   // (8, 64): 4 cols/thread, b128 loads
    for (int it = 0; it < MAX_ITER; ++it) {
        ota_col_lse_partial<<<gp, 256, 0, stream>>>(logK, f, pm, ps);
        ota_col_lse_combine<<<N_PTS / 1024, 256, 0, stream>>>(pm, ps, g);
        ota_row_lse_kernel<<<N_PTS, 256, 0, stream>>>(logK, g, f);
    }

    // 5) outputs: aligned_A then aligned_B, concatenated in d_out
    ota_apply_T_A<<<(512 * 12) / 8, 256, 0, stream>>>(logK, A, f, g, out);
    ota_apply_T_B<<<(512 * 12) / 8, 256, 0, stream>>>(logK, B, f, g,
                                                      out + (size_t)N_PTS * D_DIM);
}